// RNNModel_84121229459780
// MI455X (gfx1250) — compile-verified
//
#include <hip/hip_runtime.h>

// ---------------------------------------------------------------------------
// Stacked SimpleRNN (relu) scan on MI455X / gfx1250 using V_WMMA_F32_16X16X4_F32.
//   B=512, T=1024, STEP=64, UNITS=128.
// One workgroup per 16-row batch tile; 8 waves, wave w owns N-cols [16w,16w+16).
// Weights + ping-pong hidden state live in LDS for the whole scan.
// ---------------------------------------------------------------------------

typedef float v2f __attribute__((ext_vector_type(2)));
typedef float v8f __attribute__((ext_vector_type(8)));

#define B_TOTAL  512
#define T_STEPS  1024
#define STEP     64
#define UNITS    128
#define TILE_M   16

// LDS layout (dword units)
#define WP        288                    // pair-interleaved row-pair stride (256 data + 32 pad)
#define H_STRIDE  132                    // padded h row stride (16 rows x 128 cols + pad)
#define H_TILE_DW (TILE_M * H_STRIDE)    // 2112 dwords per h buffer

#define WX1_OFF 0                                   // 32 row-pairs * 288
#define WH1_OFF (WX1_OFF + (STEP / 2) * WP)         // 9216
#define WX2_OFF (WH1_OFF + (UNITS / 2) * WP)        // 27648
#define WH2_OFF (WX2_OFF + (UNITS / 2) * WP)        // 46080
#define H1_OFF  (WH2_OFF + (UNITS / 2) * WP)        // 64512  (2 buffers)
#define H2_OFF  (H1_OFF + 2 * H_TILE_DW)            // 68736  (2 buffers)
#define LDS_DW  (H2_OFF + 2 * H_TILE_DW)            // 72960 dwords = 291840 bytes

__global__ void __launch_bounds__(256, 1)
rnn_scan_wmma_f32(const float* __restrict__ x,
                  const float* __restrict__ Wx1, const float* __restrict__ Wh1,
                  const float* __restrict__ b1,
                  const float* __restrict__ Wx2, const float* __restrict__ Wh2,
                  const float* __restrict__ b2,
                  const float* __restrict__ Wd,  const float* __restrict__ bd,
                  float* __restrict__ out)
{
    extern __shared__ float lds[];

    const int tid   = threadIdx.x;
    const int lane  = tid & 31;
    const int wave  = tid >> 5;
    const int laneN = lane & 15;   // column within 16-wide N tile / row for A frags
    const int khalf = lane >> 4;   // which K pair of the 16x16x4 fragment
    const int n0    = wave * 16;   // this wave's N origin
    const int row0  = blockIdx.x * TILE_M;

    // ---- Stage weights into LDS, pair-interleaved so each B fragment is one b64:
    //      lds[off + (r>>1)*WP + c*2 + (r&1)] = W[r][c]
    for (int e = tid; e < STEP * UNITS; e += 256) {
        int r = e >> 7, c = e & 127;
        lds[WX1_OFF + (r >> 1) * WP + c * 2 + (r & 1)] = Wx1[e];
    }
    for (int e = tid; e < UNITS * UNITS; e += 256) {
        int r = e >> 7, c = e & 127;
        int o = (r >> 1) * WP + c * 2 + (r & 1);
        lds[WH1_OFF + o] = Wh1[e];
        lds[WX2_OFF + o] = Wx2[e];
        lds[WH2_OFF + o] = Wh2[e];
    }
    // ---- Zero initial hidden state (ping-pong buffer 0)
    for (int e = tid; e < H_TILE_DW; e += 256) {
        lds[H1_OFF + e] = 0.0f;
        lds[H2_OFF + e] = 0.0f;
    }
    const float bias1 = b1[n0 + laneN];
    const float bias2 = b2[n0 + laneN];
    __syncthreads();

    // Per-lane base of this lane's batch row in x  (x is [B][T][STEP] row-major)
    const size_t xrow = (size_t)(row0 + laneN) * (size_t)T_STEPS * STEP;

    for (int t = 0; t < T_STEPS; ++t) {
        const int cur = t & 1;
        const int nxt = cur ^ 1;
        const float* xt = x + xrow + (size_t)t * STEP;

        // Prefetch next timestep's x row (256 B = 2 cache lines)
        if (t + 1 < T_STEPS) {
            __builtin_prefetch(xt + STEP, 0, 1);
            __builtin_prefetch(xt + STEP + 32, 0, 1);
        }

        // ================= Layer 1: h1' = relu(x_t@Wx1 + h1@Wh1 + b1) ========
        v8f c1 = {};
        #pragma unroll
        for (int kb = 0; kb < STEP / 4; ++kb) {               // K = 64 from x
            v2f a = *reinterpret_cast<const v2f*>(xt + kb * 4 + khalf * 2);
            v2f b = *reinterpret_cast<const v2f*>(
                        &lds[WX1_OFF + (kb * 2 + khalf) * WP + laneN * 2]);
            c1 = __builtin_amdgcn_wmma_f32_16x16x4_f32(false, a, false, b,
                                                       (short)0, c1, false, false);
        }
        const float* h1cur = &lds[H1_OFF + cur * H_TILE_DW];
        #pragma unroll
        for (int kb = 0; kb < UNITS / 4; ++kb) {              // K = 128 from h1
            v2f a = *reinterpret_cast<const v2f*>(
                        h1cur + laneN * H_STRIDE + kb * 4 + khalf * 2);
            v2f b = *reinterpret_cast<const v2f*>(
                        &lds[WH1_OFF + (kb * 2 + khalf) * WP + laneN * 2]);
            c1 = __builtin_amdgcn_wmma_f32_16x16x4_f32(false, a, false, b,
                                                       (short)0, c1, false, false);
        }
        float* h1nxt = &lds[H1_OFF + nxt * H_TILE_DW];
        #pragma unroll
        for (int j = 0; j < 8; ++j) {                         // bias + relu + store
            float v = c1[j] + bias1;
            v = v > 0.0f ? v : 0.0f;
            h1nxt[(j + khalf * 8) * H_STRIDE + n0 + laneN] = v;
        }
        __syncthreads();   // h1' visible to all waves

        // ================= Layer 2: h2' = relu(h1'@Wx2 + h2@Wh2 + b2) ========
        v8f c2 = {};
        #pragma unroll
        for (int kb = 0; kb < UNITS / 4; ++kb) {              // K = 128 from h1'
            v2f a = *reinterpret_cast<const v2f*>(
                        h1nxt + laneN * H_STRIDE + kb * 4 + khalf * 2);
            v2f b = *reinterpret_cast<const v2f*>(
                        &lds[WX2_OFF + (kb * 2 + khalf) * WP + laneN * 2]);
            c2 = __builtin_amdgcn_wmma_f32_16x16x4_f32(false, a, false, b,
                                                       (short)0, c2, false, false);
        }
        const float* h2cur = &lds[H2_OFF + cur * H_TILE_DW];
        #pragma unroll
        for (int kb = 0; kb < UNITS / 4; ++kb) {              // K = 128 from h2
            v2f a = *reinterpret_cast<const v2f*>(
                        h2cur + laneN * H_STRIDE + kb * 4 + khalf * 2);
            v2f b = *reinterpret_cast<const v2f*>(
                        &lds[WH2_OFF + (kb * 2 + khalf) * WP + laneN * 2]);
            c2 = __builtin_amdgcn_wmma_f32_16x16x4_f32(false, a, false, b,
                                                       (short)0, c2, false, false);
        }
        float* h2nxt = &lds[H2_OFF + nxt * H_TILE_DW];
        #pragma unroll
        for (int j = 0; j < 8; ++j) {
            float v = c2[j] + bias2;
            v = v > 0.0f ? v : 0.0f;
            h2nxt[(j + khalf * 8) * H_STRIDE + n0 + laneN] = v;
        }
        __syncthreads();   // h2' visible for next step
    }

    // ---- Head: out[b] = h2_final . Wd + bd   (final buffer = ((T-1)&1)^1)
    const int fin = ((T_STEPS - 1) & 1) ^ 1;
    if (tid < TILE_M) {
        const float* h2f = &lds[H2_OFF + fin * H_TILE_DW + tid * H_STRIDE];
        float acc = bd[0];
        #pragma unroll 4
        for (int k = 0; k < UNITS; ++k) acc += h2f[k] * Wd[k];
        out[row0 + tid] = acc;
    }
}

extern "C" void kernel_launch(void* const* d_in, const int* in_sizes, int n_in,
                              void* d_out, int out_size, void* d_ws, size_t ws_size,
                              hipStream_t stream) {
    (void)in_sizes; (void)n_in; (void)d_ws; (void)ws_size; (void)out_size;
    const float* x   = (const float*)d_in[0];
    const float* Wx1 = (const float*)d_in[1];
    const float* Wh1 = (const float*)d_in[2];
    const float* b1  = (const float*)d_in[3];
    const float* Wx2 = (const float*)d_in[4];
    const float* Wh2 = (const float*)d_in[5];
    const float* b2  = (const float*)d_in[6];
    const float* Wd  = (const float*)d_in[7];
    const float* bd  = (const float*)d_in[8];
    float* out = (float*)d_out;

    const size_t shmem = (size_t)LDS_DW * sizeof(float);   // 291,840 B < 320 KB WGP LDS
    hipFuncSetAttribute((const void*)rnn_scan_wmma_f32,
                        hipFuncAttributeMaxDynamicSharedMemorySize, (int)shmem);

    rnn_scan_wmma_f32<<<B_TOTAL / TILE_M, 256, shmem, stream>>>(
        x, Wx1, Wh1, b1, Wx2, Wh2, b2, Wd, bd, out);
}